// SimpleGridGNN_78460462563808
// MI455X (gfx1250) — compile-verified
//
#include <hip/hip_runtime.h>

// Problem dims (fixed by the reference)
#define N_B   16
#define V_N   16384
#define D_F   128
#define E_N   65536
#define L_N   2
#define EPSV  1e-5f

#define M_TOTAL (N_B * V_N)          // 262144 rows in the flattened GEMM
#define CHUNK_ROWS 1024
#define N_CHUNKS (M_TOTAL / CHUNK_ROWS)  // 256

typedef __attribute__((ext_vector_type(2))) float v2f;
typedef __attribute__((ext_vector_type(8))) float v8f;

// ---------------------------------------------------------------------------
// Stage 1: SpMM scatter.  One wave32 per (batch n, edge e); each lane owns 4
// contiguous features.  float4 gather from X[n,src], fp32 atomic adds into
// X_nei[n,dst].  Per-batch node slab is 8 MB -> lives in the 192 MB L2, so
// HBM sees ~one pass over X despite the random edge pattern.
// ---------------------------------------------------------------------------
__global__ void spmm_scatter(const float* __restrict__ X,
                             const int*   __restrict__ esrc,
                             const int*   __restrict__ edst,
                             const float* __restrict__ avals,
                             float*       __restrict__ Xn) {
  unsigned gtid = blockIdx.x * blockDim.x + threadIdx.x;
  unsigned wid  = gtid >> 5;
  unsigned lane = threadIdx.x & 31;
  if (wid >= (unsigned)(N_B * E_N)) return;
  unsigned n = wid / E_N;
  unsigned e = wid % E_N;
  int   s = esrc[e];
  int   d = edst[e];
  float a = avals[e];
  const float* xp = X  + ((size_t)n * V_N + (size_t)s) * D_F + lane * 4u;
  float*       yp = Xn + ((size_t)n * V_N + (size_t)d) * D_F + lane * 4u;
  float4 v = *(const float4*)xp;
  atomicAdd(yp + 0, a * v.x);
  atomicAdd(yp + 1, a * v.y);
  atomicAdd(yp + 2, a * v.z);
  atomicAdd(yp + 3, a * v.w);
}

// ---------------------------------------------------------------------------
// Stage 2: dual GEMM  Y = X*Ws^T + Xn*Wn^T  with V_WMMA_F32_16X16X4_F32.
// One wave per 16x16 output tile; K=128 -> 32 k-steps, 2 WMMAs per step
// accumulating into a single f32 C tile.  Grid exactly covers all tiles
// (no divergence: EXEC must be all-ones for WMMA).
// A layout: lane%16 = row, K = v + 2*(lane/16) + 4*kk  -> 8-byte load
// B layout: lane%16 = col, same K mapping -> row-of-W 8-byte load (B = W^T).
// ---------------------------------------------------------------------------
__global__ void gemm_dual_wmma(const float* __restrict__ X,
                               const float* __restrict__ Xn,
                               const float* __restrict__ Ws,
                               const float* __restrict__ Wn,
                               float*       __restrict__ Y) {
  const int lane = threadIdx.x & 31;
  const int wave = threadIdx.x >> 5;                  // blockDim.x == 128
  const int tile = blockIdx.x * 4 + wave;
  const int tn   = tile & 7;                          // D_F/16 == 8 col tiles
  const int tm   = tile >> 3;
  const int half = lane >> 4;
  const int lrow = lane & 15;

  const size_t m    = (size_t)tm * 16 + (size_t)lrow;  // A row (flattened n*V+v)
  const size_t nout = (size_t)tn * 16 + (size_t)lrow;  // B column = output feat

  const float* xr  = X  + m * D_F;
  const float* xnr = Xn + m * D_F;
  const float* wsr = Ws + nout * D_F;
  const float* wnr = Wn + nout * D_F;

  v8f c = {};
#pragma unroll 8
  for (int kk = 0; kk < D_F / 4; ++kk) {
    const int k0 = kk * 4 + half * 2;                 // even -> 8B aligned
    v2f ax = *(const v2f*)(xr  + k0);
    v2f an = *(const v2f*)(xnr + k0);
    v2f bs = *(const v2f*)(wsr + k0);
    v2f bn = *(const v2f*)(wnr + k0);
    c = __builtin_amdgcn_wmma_f32_16x16x4_f32(false, ax, false, bs,
                                              (short)0, c, false, false);
    c = __builtin_amdgcn_wmma_f32_16x16x4_f32(false, an, false, bn,
                                              (short)0, c, false, false);
  }

  // C/D layout: VGPR r, lane-half h -> row r + 8h, col lane%16
  float* yb = Y + ((size_t)tm * 16) * D_F + (size_t)tn * 16;
#pragma unroll
  for (int r = 0; r < 8; ++r) {
    yb[(size_t)(r + 8 * half) * D_F + lrow] = c[r];
  }
}

// ---------------------------------------------------------------------------
// Stage 3a: BatchNorm partial reduction.  128 threads per block (thread d
// owns feature d -> coalesced), each block reduces a fixed 1024-row chunk in
// a fixed order -> deterministic.
// ---------------------------------------------------------------------------
__global__ void bn_partial(const float* __restrict__ Y,
                           float*       __restrict__ part) {
  const int d     = threadIdx.x;     // 0..127
  const int chunk = blockIdx.x;      // 0..255
  const float* p = Y + (size_t)chunk * CHUNK_ROWS * D_F + d;
  float s = 0.f, s2 = 0.f;
  for (int r = 0; r < CHUNK_ROWS; ++r) {
    float v = p[(size_t)r * D_F];
    s  += v;
    s2 += v * v;
  }
  part[(size_t)chunk * (2 * D_F) + d]       = s;
  part[(size_t)chunk * (2 * D_F) + D_F + d] = s2;
}

// Stage 3b: finalize -> per-feature scale/shift (single block, fixed order).
__global__ void bn_finalize(const float* __restrict__ part,
                            const float* __restrict__ gamma,
                            const float* __restrict__ beta,
                            float*       __restrict__ stats) {
  const int d = threadIdx.x;         // 0..127
  float s = 0.f, s2 = 0.f;
  for (int c = 0; c < N_CHUNKS; ++c) {
    s  += part[(size_t)c * (2 * D_F) + d];
    s2 += part[(size_t)c * (2 * D_F) + D_F + d];
  }
  const float inv  = 1.0f / (float)M_TOTAL;
  const float mean = s * inv;
  const float var  = fmaxf(s2 * inv - mean * mean, 0.0f);  // biased variance
  const float sc   = gamma[d] * rsqrtf(var + EPSV);
  stats[d]       = sc;
  stats[D_F + d] = beta[d] - mean * sc;
}

// Stage 4: fused normalize + ReLU, elementwise.
__global__ void bn_apply_relu(const float* __restrict__ Y,
                              const float* __restrict__ stats,
                              float*       __restrict__ Xout) {
  size_t i = (size_t)blockIdx.x * blockDim.x + threadIdx.x;
  int d = (int)(i & (D_F - 1));
  float v = fmaf(Y[i], stats[d], stats[D_F + d]);
  Xout[i] = fmaxf(v, 0.0f);
}

// ---------------------------------------------------------------------------
// Host-side orchestration.  Workspace layout:
//   [0, NVD)            : X_nei
//   [NVD, 2*NVD)        : Y
//   [2*NVD, +2*256*128) : BN partials
//   then 256 floats     : scale/shift
// Layer 1 activation lands in d_out; layer 2 reads it and overwrites d_out.
// ---------------------------------------------------------------------------
extern "C" void kernel_launch(void* const* d_in, const int* in_sizes, int n_in,
                              void* d_out, int out_size, void* d_ws, size_t ws_size,
                              hipStream_t stream) {
  const float* H     = (const float*)d_in[0];
  const int*   esrc  = (const int*)  d_in[1];
  const int*   edst  = (const int*)  d_in[2];
  const float* avals = (const float*)d_in[3];
  const float* Wself = (const float*)d_in[4];  // [L, D, D]
  const float* Wnei  = (const float*)d_in[5];  // [L, D, D]
  const float* gamma = (const float*)d_in[6];  // [L, D]
  const float* beta  = (const float*)d_in[7];  // [L, D]
  float* out = (float*)d_out;

  const size_t NVD = (size_t)N_B * V_N * D_F;
  float* Xn    = (float*)d_ws;
  float* Y     = Xn + NVD;
  float* part  = Y + NVD;
  float* stats = part + (size_t)N_CHUNKS * 2 * D_F;

  for (int l = 0; l < L_N; ++l) {
    const float* X = (l == 0) ? H : out;

    hipMemsetAsync(Xn, 0, NVD * sizeof(float), stream);

    {
      const long long threads_total = (long long)N_B * E_N * 32;  // 1 wave/(n,e)
      const int tpb = 256;
      const int blocks = (int)((threads_total + tpb - 1) / tpb);
      spmm_scatter<<<blocks, tpb, 0, stream>>>(X, esrc, edst, avals, Xn);
    }
    {
      const int tiles = (M_TOTAL / 16) * (D_F / 16);  // 131072
      gemm_dual_wmma<<<tiles / 4, 128, 0, stream>>>(
          X, Xn,
          Wself + (size_t)l * D_F * D_F,
          Wnei  + (size_t)l * D_F * D_F,
          Y);
    }
    bn_partial<<<N_CHUNKS, D_F, 0, stream>>>(Y, part);
    bn_finalize<<<1, D_F, 0, stream>>>(part, gamma + (size_t)l * D_F,
                                       beta + (size_t)l * D_F, stats);
    bn_apply_relu<<<(int)(NVD / 256), 256, 0, stream>>>(Y, stats, out);
  }
}